// YOLOv3_25314537243282
// MI455X (gfx1250) — compile-verified
//
#include <hip/hip_runtime.h>
#include <cstring>

// ---------------- constants ----------------
#define NBINS    65536      // score-bit histogram bins (bits [30:15])
#define CAND_CAP 4096       // per-level candidate cap (LDS bitonic size)
#define K_LEVEL  1000
#define K_TOTAL  3000
#define NSORT    4096       // global sort size (3000 padded)
#define NCLS     80

// float4-groups of class scores per level: rows * (80/4)
#define G0 1536000          // 76800*20
#define G1 384000           // 19200*20
#define G2 96000            //  4800*20
#define GTOT 2016000

// workspace byte offsets
#define OFF_HIST 0u                   // 3*65536*4 = 786432
#define OFF_CNT  786432u              // 3 u32
#define OFF_THR  786448u              // 3 u32
#define OFF_CAND 786464u              // 3*4096*8 = 98304
#define OFF_TOP  884768u              // 3000*8   = 24000
#define OFF_FBOX 908768u              // 3000*4*4 = 48000
#define OFF_FSC  956768u              // 3000*4
#define OFF_FLB  968768u              // 3000*4
#define ZERO_U32 196616u              // zero hist+cnt+thr

__constant__ float d_anch[3][3][2] = {
  {{10.f,13.f},{16.f,30.f},{33.f,23.f}},
  {{30.f,61.f},{62.f,45.f},{59.f,119.f}},
  {{116.f,90.f},{156.f,198.f},{373.f,326.f}}};

__device__ __forceinline__ float sigf(float x) { return 1.0f / (1.0f + __expf(-x)); }

// ---- CDNA5 async global->LDS copy (gfx1250 VGLOBAL async path, ASYNCcnt) ----
__device__ __forceinline__ void async_copy_b64(void* lds_dst, const void* gsrc) {
  unsigned l = (unsigned)(unsigned long long)lds_dst;     // low 32 bits = LDS offset
  unsigned long long ga = (unsigned long long)gsrc;
  asm volatile("global_load_async_to_lds_b64 %0, %1, off"
               :: "v"(l), "v"(ga) : "memory");
}
__device__ __forceinline__ void async_wait0() {
  asm volatile("s_wait_asynccnt 0" ::: "memory");
}

// ---------------- kernels ----------------
__global__ void zero_kernel(unsigned* p, int n) {
  int i = blockIdx.x * blockDim.x + threadIdx.x;
  if (i < n) p[i] = 0u;
}

__device__ __forceinline__ void map_group(long long g, int& level, long long& lg) {
  if (g < G0)            { level = 0; lg = g; }
  else if (g < G0 + G1)  { level = 1; lg = g - G0; }
  else                   { level = 2; lg = g - (G0 + G1); }
}

__global__ void hist_kernel(const float* __restrict__ obj0, const float* __restrict__ cls0,
                            const float* __restrict__ obj1, const float* __restrict__ cls1,
                            const float* __restrict__ obj2, const float* __restrict__ cls2,
                            unsigned* __restrict__ hist, unsigned minKey) {
  long long g = (long long)blockIdx.x * blockDim.x + threadIdx.x;
  if (g >= GTOT) return;
  int level; long long lg; map_group(g, level, lg);
  const float* obj = level == 0 ? obj0 : level == 1 ? obj1 : obj2;
  const float* cls = level == 0 ? cls0 : level == 1 ? cls1 : cls2;
  int a = (int)(lg / 20), cg = (int)(lg % 20);
  const float* row = cls + (long long)a * NCLS;
  __builtin_prefetch(row + 256);                    // global_prefetch_b8
  float so = sigf(obj[a]);
  float4 cv = ((const float4*)row)[cg];
  unsigned* h = hist + level * NBINS;
  float c4[4] = {cv.x, cv.y, cv.z, cv.w};
#pragma unroll
  for (int i = 0; i < 4; ++i) {
    unsigned key = __float_as_uint(so * sigf(c4[i]));
    if (key >= minKey) atomicAdd(&h[key >> 15], 1u);
  }
}

__global__ void threshold_kernel(const unsigned* __restrict__ hist,
                                 unsigned* __restrict__ thrBin, unsigned minKey) {
  __shared__ unsigned scan[1024];
  __shared__ unsigned sRun;
  __shared__ int sFound;
  const int level = blockIdx.x;
  const unsigned* h = hist + level * NBINS;
  const int minBin = (int)(minKey >> 15);
  if (threadIdx.x == 0) { sRun = 0u; sFound = -1; }
  __syncthreads();
  for (int it = 0; it < NBINS / 1024; ++it) {
    int pos = it * 1024 + (int)threadIdx.x;        // descending-bin scan position
    int bin = NBINS - 1 - pos;
    unsigned v = (bin >= minBin) ? h[bin] : 0u;
    scan[threadIdx.x] = v;
    __syncthreads();
    for (int off = 1; off < 1024; off <<= 1) {     // Hillis-Steele inclusive scan
      unsigned t = (threadIdx.x >= (unsigned)off) ? scan[threadIdx.x - off] : 0u;
      __syncthreads();
      scan[threadIdx.x] += t;
      __syncthreads();
    }
    unsigned run = sRun;
    unsigned cum  = run + scan[threadIdx.x];
    unsigned prev = run + (threadIdx.x ? scan[threadIdx.x - 1] : 0u);
    if (prev < (unsigned)K_LEVEL && cum >= (unsigned)K_LEVEL) sFound = bin;  // unique
    __syncthreads();
    if (threadIdx.x == 1023) sRun = cum;
    __syncthreads();
    if (sFound >= 0) break;
    if (NBINS - 1 - (it * 1024 + 1023) <= minBin) break;   // uniform
  }
  __syncthreads();
  if (threadIdx.x == 0) {
    int t = sFound;
    if (t < 0 || t < minBin) t = minBin;
    thrBin[level] = (unsigned)t;
  }
}

__global__ void compact_kernel(const float* __restrict__ obj0, const float* __restrict__ cls0,
                               const float* __restrict__ obj1, const float* __restrict__ cls1,
                               const float* __restrict__ obj2, const float* __restrict__ cls2,
                               const unsigned* __restrict__ thrBin,
                               unsigned* __restrict__ candCount, uint2* __restrict__ cand,
                               unsigned minKey) {
  long long g = (long long)blockIdx.x * blockDim.x + threadIdx.x;
  if (g >= GTOT) return;
  int level; long long lg; map_group(g, level, lg);
  const float* obj = level == 0 ? obj0 : level == 1 ? obj1 : obj2;
  const float* cls = level == 0 ? cls0 : level == 1 ? cls1 : cls2;
  int a = (int)(lg / 20), cg = (int)(lg % 20);
  unsigned tk = thrBin[level] << 15;
  if (tk < minKey) tk = minKey;
  float so = sigf(obj[a]);
  float4 cv = ((const float4*)(cls + (long long)a * NCLS))[cg];
  float c4[4] = {cv.x, cv.y, cv.z, cv.w};
#pragma unroll
  for (int i = 0; i < 4; ++i) {
    unsigned key = __float_as_uint(so * sigf(c4[i]));
    if (key >= tk) {
      unsigned pos = atomicAdd(&candCount[level], 1u);
      if (pos < CAND_CAP) {
        uint2 e; e.x = key; e.y = (unsigned)(a * NCLS + cg * 4 + i);
        cand[level * CAND_CAP + pos] = e;
      }
    }
  }
}

// ascending bitonic sort of N u64 keys in LDS, blockDim.x == 1024
template <int N>
__device__ void bitonic_keys(unsigned long long* k) {
  for (int sz = 2; sz <= N; sz <<= 1)
    for (int st = sz >> 1; st > 0; st >>= 1) {
      __syncthreads();
      for (int i = threadIdx.x; i < N; i += 1024) {
        int j = i ^ st;
        if (j > i) {
          bool up = ((i & sz) == 0);
          unsigned long long a = k[i], b = k[j];
          if ((a > b) == up) { k[i] = b; k[j] = a; }
        }
      }
    }
  __syncthreads();
}

template <int N>
__device__ void bitonic_kv(unsigned long long* k, unsigned* v) {
  for (int sz = 2; sz <= N; sz <<= 1)
    for (int st = sz >> 1; st > 0; st >>= 1) {
      __syncthreads();
      for (int i = threadIdx.x; i < N; i += 1024) {
        int j = i ^ st;
        if (j > i) {
          bool up = ((i & sz) == 0);
          unsigned long long a = k[i], b = k[j];
          if ((a > b) == up) {
            k[i] = b; k[j] = a;
            unsigned t = v[i]; v[i] = v[j]; v[j] = t;
          }
        }
      }
    }
  __syncthreads();
}

// one block per level: sort candidates desc by score (tie: lowest idx), emit top 1000
__global__ void level_sort_kernel(const unsigned* __restrict__ candCount,
                                  const uint2* __restrict__ cand,
                                  uint2* __restrict__ levelTop) {
  __shared__ unsigned long long keys[CAND_CAP];
  const int level = blockIdx.x;
  unsigned cnt = candCount[level];
  if (cnt > CAND_CAP) cnt = CAND_CAP;
  const uint2* c = cand + level * CAND_CAP;
  for (int s = threadIdx.x; s < CAND_CAP; s += 1024) {
    if (s < (int)cnt) async_copy_b64(&keys[s], &c[s]);     // {scoreBits, idx} raw
    else keys[s] = ~0ULL;
  }
  async_wait0();
  __syncthreads();
  for (int s = threadIdx.x; s < (int)cnt; s += 1024) {
    unsigned long long raw = keys[s];
    unsigned sb = (unsigned)raw;            // low dword  = scoreBits
    unsigned ix = (unsigned)(raw >> 32);    // high dword = idx
    keys[s] = ((unsigned long long)(~sb) << 32) | ix;   // ascending == (score desc, idx asc)
  }
  __syncthreads();
  bitonic_keys<CAND_CAP>(keys);
  for (int r = threadIdx.x; r < K_LEVEL; r += 1024) {
    uint2 out;
    if (r < (int)cnt) {
      unsigned long long k = keys[r];
      out.x = ~(unsigned)(k >> 32);         // scoreBits (p = sigo*sigc)
      out.y = (unsigned)k;                  // idx
    } else { out.x = 0u; out.y = 0u; }      // score 0 pad -> below conf -> zero output
    levelTop[level * K_LEVEL + r] = out;
  }
}

// one block: global top-3000 sort (score desc, concat-slot tiebreak) + box decode
__global__ void global_sort_decode_kernel(const uint2* __restrict__ levelTop,
                                          const float* __restrict__ reg0,
                                          const float* __restrict__ reg1,
                                          const float* __restrict__ reg2,
                                          float* __restrict__ fbox,
                                          float* __restrict__ fscore,
                                          int* __restrict__ flabel) {
  __shared__ unsigned long long keys[NSORT];
  __shared__ unsigned vals[NSORT];
  for (int s = threadIdx.x; s < NSORT; s += 1024) {
    if (s < K_TOTAL) async_copy_b64(&keys[s], &levelTop[s]);
    else { keys[s] = ~0ULL; vals[s] = 0u; }
  }
  async_wait0();
  __syncthreads();
  for (int s = threadIdx.x; s < K_TOTAL; s += 1024) {
    unsigned long long raw = keys[s];
    unsigned sb = (unsigned)raw;
    unsigned ix = (unsigned)(raw >> 32);          // a*80+c < 2^23
    int level = s / K_LEVEL;
    keys[s] = ((unsigned long long)(~sb) << 32) | (unsigned)s;
    vals[s] = ((unsigned)level << 24) | ix;
  }
  __syncthreads();
  bitonic_kv<NSORT>(keys, vals);
  for (int r = threadIdx.x; r < K_TOTAL; r += 1024) {
    unsigned sb = ~(unsigned)(keys[r] >> 32);
    float p = __uint_as_float(sb);
    unsigned v = vals[r];
    int level = (int)(v >> 24);
    int idx = (int)(v & 0xFFFFFFu);
    int label = idx % NCLS;
    int a = idx / NCLS;
    const float* reg = level == 0 ? reg0 : level == 1 ? reg1 : reg2;
    int stride = 8 << level;
    int f = 1280 / stride;
    int cell = a / 3, b = a % 3;
    float gx = (float)(cell % f), gy = (float)(cell / f);
    float4 rg = ((const float4*)reg)[a];
    float cx = (sigf(rg.x) + gx) * (float)stride;
    float cy = (sigf(rg.y) + gy) * (float)stride;
    float w = __expf(rg.z) * d_anch[level][b][0];
    float h = __expf(rg.w) * d_anch[level][b][1];
    fbox[4 * r + 0] = cx - w * 0.5f;
    fbox[4 * r + 1] = cy - h * 0.5f;
    fbox[4 * r + 2] = cx + w * 0.5f;
    fbox[4 * r + 3] = cy + h * 0.5f;
    fscore[r] = sqrtf(p);
    flabel[r] = label;
  }
}

// one block: greedy NMS exactly mirroring the reference's (ctr,wh) reinterpretation
__global__ void nms_kernel(const float* __restrict__ fbox, const float* __restrict__ fscore,
                           const int* __restrict__ flabel, float* __restrict__ out) {
  __shared__ float X1[K_TOTAL], Y1[K_TOTAL], X2[K_TOTAL], Y2[K_TOTAL], AR[K_TOTAL];
  __shared__ unsigned char KP[K_TOTAL];
  for (int j = threadIdx.x; j < K_TOTAL; j += 1024) {
    float b0 = fbox[4 * j + 0], b1 = fbox[4 * j + 1];
    float b2 = fbox[4 * j + 2], b3 = fbox[4 * j + 3];
    float off = (float)flabel[j] * 10000000.0f;     // CLASS_OFFSET, fp32 like the ref
    float x1 = (b0 - b2 * 0.5f) + off;
    float y1 = b1 - b3 * 0.5f;
    float x2 = (b0 + b2 * 0.5f) + off;
    float y2 = b1 + b3 * 0.5f;
    X1[j] = x1; Y1[j] = y1; X2[j] = x2; Y2[j] = y2;
    AR[j] = (x2 - x1) * (y2 - y1);
    KP[j] = (fscore[j] > 0.3f) ? 1 : 0;
  }
  __syncthreads();
  for (int i = 0; i < K_TOTAL; ++i) {
    if (KP[i]) {
      float xi1 = X1[i], yi1 = Y1[i], xi2 = X2[i], yi2 = Y2[i], ai = AR[i];
      for (int j = i + 1 + (int)threadIdx.x; j < K_TOTAL; j += 1024) {
        if (KP[j]) {
          float xx1 = fmaxf(xi1, X1[j]), yy1 = fmaxf(yi1, Y1[j]);
          float xx2 = fminf(xi2, X2[j]), yy2 = fminf(yi2, Y2[j]);
          float inter = fmaxf(1e-10f, xx2 - xx1) * fmaxf(1e-10f, yy2 - yy1);
          float iou = inter / (ai + AR[j] - inter + 1e-14f);
          if (iou > 0.5f) KP[j] = 0;
        }
      }
    }
    __syncthreads();
  }
  // outputs: boxes[12000] | scores[3000] | labels[3000] | keep[3000]
  for (int r = threadIdx.x; r < K_TOTAL; r += 1024) {
    bool k = KP[r] != 0;
    out[4 * r + 0] = k ? fbox[4 * r + 0] : 0.0f;
    out[4 * r + 1] = k ? fbox[4 * r + 1] : 0.0f;
    out[4 * r + 2] = k ? fbox[4 * r + 2] : 0.0f;
    out[4 * r + 3] = k ? fbox[4 * r + 3] : 0.0f;
    out[12000 + r] = k ? fscore[r] : 0.0f;
    out[15000 + r] = k ? (float)flabel[r] : -1.0f;
    out[18000 + r] = k ? 1.0f : 0.0f;
  }
}

// ---------------- host launcher ----------------
extern "C" void kernel_launch(void* const* d_in, const int* in_sizes, int n_in,
                              void* d_out, int out_size, void* d_ws, size_t ws_size,
                              hipStream_t stream) {
  const float* obj0 = (const float*)d_in[0];
  const float* cls0 = (const float*)d_in[1];
  const float* reg0 = (const float*)d_in[2];
  const float* obj1 = (const float*)d_in[3];
  const float* cls1 = (const float*)d_in[4];
  const float* reg1 = (const float*)d_in[5];
  const float* obj2 = (const float*)d_in[6];
  const float* cls2 = (const float*)d_in[7];
  const float* reg2 = (const float*)d_in[8];

  char* ws = (char*)d_ws;
  unsigned* hist      = (unsigned*)(ws + OFF_HIST);
  unsigned* candCount = (unsigned*)(ws + OFF_CNT);
  unsigned* thrBin    = (unsigned*)(ws + OFF_THR);
  uint2*    cand      = (uint2*)   (ws + OFF_CAND);
  uint2*    levelTop  = (uint2*)   (ws + OFF_TOP);
  float*    fbox      = (float*)   (ws + OFF_FBOX);
  float*    fscore    = (float*)   (ws + OFF_FSC);
  int*      flabel    = (int*)     (ws + OFF_FLB);

  float mk = 0.09f;                 // p threshold: score>0.3  <=>  p>0.09
  unsigned minKey;
  std::memcpy(&minKey, &mk, 4);

  zero_kernel<<<(ZERO_U32 + 255) / 256, 256, 0, stream>>>(hist, (int)ZERO_U32);
  hist_kernel<<<(GTOT + 255) / 256, 256, 0, stream>>>(obj0, cls0, obj1, cls1, obj2, cls2,
                                                      hist, minKey);
  threshold_kernel<<<3, 1024, 0, stream>>>(hist, thrBin, minKey);
  compact_kernel<<<(GTOT + 255) / 256, 256, 0, stream>>>(obj0, cls0, obj1, cls1, obj2, cls2,
                                                         thrBin, candCount, cand, minKey);
  level_sort_kernel<<<3, 1024, 0, stream>>>(candCount, cand, levelTop);
  global_sort_decode_kernel<<<1, 1024, 0, stream>>>(levelTop, reg0, reg1, reg2,
                                                    fbox, fscore, flabel);
  nms_kernel<<<1, 1024, 0, stream>>>(fbox, fscore, flabel, (float*)d_out);
}